// WeightGRU_54717883351305
// MI455X (gfx1250) — compile-verified
//
#include <hip/hip_runtime.h>
#include <hip/hip_bf16.h>

// ---------------------------------------------------------------------------
// GRU autoencoder, 4096 sequential steps.
// Algebraic fold: for t>=1,  ig_t = A h_t + u  with A = W_ih (W_enc W_dec),
// removing the 8192-dim from the sequential path. Phases:
//   K1: e0 = W_enc x0 + b_enc ; c = W_enc b_dec + b_enc       (matvec, VALU)
//   K2: M  = W_enc W_dec  (128x128, K=8192)                   (fp32 WMMA)
//   K3: A = W_ih M ; u = W_ih c + bias ; ig0 = W_ih e0 + bias (small)
//   K4: sequential recurrence, 1 workgroup, weights in VGPRs  (latency path)
//   K5: out = H W_dec^T + b_dec  (4096x8192)                  (fp32 WMMA)
// ---------------------------------------------------------------------------

#define INPUT_DIM 8192
#define HIDDEN    128
#define STEPS     4096
#define GATES     384   // 3*HIDDEN
#define ROWS      768   // 6*HIDDEN  ([A ; W_hh] stacked)

typedef float v2f __attribute__((ext_vector_type(2)));
typedef float v8f __attribute__((ext_vector_type(8)));

// ------------------------- K1: encoder matvecs -----------------------------
// e0[r] = dot(W_enc[r,:], x0) + b_enc[r] ; c[r] = dot(W_enc[r,:], b_dec) + b_enc[r]
__global__ void __launch_bounds__(256) k_enc_vec(
    const float* __restrict__ Wenc, const float* __restrict__ benc,
    const float* __restrict__ x0,   const float* __restrict__ bdec,
    float* __restrict__ e0, float* __restrict__ c)
{
    const int r = blockIdx.x;
    const int t = threadIdx.x;
    const float* w = Wenc + (size_t)r * INPUT_DIM;
    float a0 = 0.f, a1 = 0.f;
    for (int i = t; i < INPUT_DIM; i += 256) {
        float wv = w[i];
        a0 = fmaf(wv, x0[i], a0);
        a1 = fmaf(wv, bdec[i], a1);
    }
    __shared__ float s0[256], s1[256];
    s0[t] = a0; s1[t] = a1;
    __syncthreads();
    for (int off = 128; off > 0; off >>= 1) {
        if (t < off) { s0[t] += s0[t + off]; s1[t] += s1[t + off]; }
        __syncthreads();
    }
    if (t == 0) { e0[r] = s0[0] + benc[r]; c[r] = s1[0] + benc[r]; }
}

// ------------------------- K2: M = W_enc @ W_dec (WMMA f32) ----------------
// 8x8 grid of 16x16 tiles, one wave per tile, K=8192 in steps of 4.
__global__ void __launch_bounds__(32) k_gemm_M_wmma(
    const float* __restrict__ Wenc, const float* __restrict__ Wdec,
    float* __restrict__ M)
{
    const int tile = blockIdx.x;          // 0..63
    const int tm = (tile >> 3) * 16;      // row tile base in M
    const int tn = (tile & 7) * 16;       // col tile base in M
    const int l  = threadIdx.x;           // 0..31
    const int mr = l & 15;                // A row / B col index
    const int kg = (l >> 4) * 2;          // K group: lanes 0-15 -> K 0,1; 16-31 -> K 2,3

    v8f acc = {};
    for (int kb = 0; kb < INPUT_DIM; kb += 4) {
        // A tile: W_enc[tm+mr, kb+kg+{0,1}] -- contiguous pair, 8B aligned
        v2f a = *(const v2f*)(Wenc + (size_t)(tm + mr) * INPUT_DIM + kb + kg);
        // B tile: B[k][n] = W_dec[kb+kg+j, tn+mr] -- stride 128, two b32 loads
        v2f b;
        b.x = Wdec[(size_t)(kb + kg + 0) * HIDDEN + tn + mr];
        b.y = Wdec[(size_t)(kb + kg + 1) * HIDDEN + tn + mr];
        acc = __builtin_amdgcn_wmma_f32_16x16x4_f32(false, a, false, b,
                                                    (short)0, acc, false, false);
    }
    const int nn = l & 15;
    const int mh = (l >> 4) * 8;
    #pragma unroll
    for (int v = 0; v < 8; ++v)
        M[(size_t)(tm + mh + v) * HIDDEN + tn + nn] = acc[v];
}

// ------------------------- K3: fold through W_ih ---------------------------
// A[r,:] = W_ih[r,:] @ M ; u[r] = dot(W_ih[r,:], c)+bias[r] ; ig0 likewise w/ e0
__global__ void __launch_bounds__(128) k_fold(
    const float* __restrict__ Wih, const float* __restrict__ bias,
    const float* __restrict__ M,   const float* __restrict__ c,
    const float* __restrict__ e0,
    float* __restrict__ A, float* __restrict__ u, float* __restrict__ ig0)
{
    const int r = blockIdx.x;     // 0..383
    const int j = threadIdx.x;    // 0..127
    __shared__ float wrow[HIDDEN];
    __shared__ float red0[HIDDEN], red1[HIDDEN];
    wrow[j] = Wih[r * HIDDEN + j];
    __syncthreads();
    float acc = 0.f;
    for (int k = 0; k < HIDDEN; ++k)
        acc = fmaf(wrow[k], M[k * HIDDEN + j], acc);
    A[r * HIDDEN + j] = acc;

    red0[j] = wrow[j] * c[j];
    red1[j] = wrow[j] * e0[j];
    __syncthreads();
    for (int off = 64; off > 0; off >>= 1) {
        if (j < off) { red0[j] += red0[j + off]; red1[j] += red1[j + off]; }
        __syncthreads();
    }
    if (j == 0) { u[r] = red0[0] + bias[r]; ig0[r] = red1[0] + bias[r]; }
}

// ------------------------- K4: sequential GRU recurrence -------------------
// One workgroup (24 wave32 waves) on one WGP. Each lane owns one row of
// [A ; W_hh] resident in 128 f32 VGPRs (persistent-RNN). h broadcast via LDS.
__global__ void __launch_bounds__(ROWS, 1) k_gru_recur(
    const float* __restrict__ A, const float* __restrict__ Whh,
    const float* __restrict__ u, const float* __restrict__ ig0,
    const float* __restrict__ bias_n, float* __restrict__ Hs)
{
    const int row = threadIdx.x;                 // 0..767
    __shared__ __align__(16) float hs[HIDDEN];   // current h (f32)
    __shared__ float y[ROWS];                    // gate pre-activations

    // Load this lane's weight row into registers (fully unrolled -> VGPRs).
    const float* wp = (row < GATES) ? (A + row * HIDDEN)
                                    : (Whh + (row - GATES) * HIDDEN);
    float w[HIDDEN];
    #pragma unroll
    for (int k = 0; k < HIDDEN; ++k) w[k] = wp[k];

    const float urow = (row < GATES) ? u[row]   : 0.f;
    const float i0v  = (row < GATES) ? ig0[row] : 0.f;
    float bn = 0.f, hreg = 0.f;
    if (row < HIDDEN) bn = bias_n[row];

    for (int t = 0; t < STEPS; ++t) {
        float acc;
        if (t == 0) {
            acc = (row < GATES) ? i0v : 0.f;     // h_0 = 0  =>  hg = 0
        } else {
            acc = urow;
            const float4* h4 = (const float4*)hs;
            #pragma unroll
            for (int k = 0; k < HIDDEN / 4; ++k) {
                float4 hv = h4[k];               // ds_load_b128 broadcast
                acc = fmaf(w[4 * k + 0], hv.x, acc);
                acc = fmaf(w[4 * k + 1], hv.y, acc);
                acc = fmaf(w[4 * k + 2], hv.z, acc);
                acc = fmaf(w[4 * k + 3], hv.w, acc);
            }
        }
        y[row] = acc;
        __syncthreads();                          // y ready; hs reads done

        if (row < HIDDEN) {
            float ir = y[row],              iz = y[row + HIDDEN];
            float inn = y[row + 2*HIDDEN],  hr = y[row + 3*HIDDEN];
            float hz = y[row + 4*HIDDEN],   hn = y[row + 5*HIDDEN];
            float r = 1.f / (1.f + expf(-(ir + hr)));
            float z = 1.f / (1.f + expf(-(iz + hz)));
            float n = tanhf(inn + r * (hn + bn));
            hreg = n + z * (hreg - n);            // equinox GRU update
            hs[row] = hreg;
            Hs[t * HIDDEN + row] = hreg;          // h_{t+1} -> row t
        }
        __syncthreads();                          // hs updated for next step
    }
}

// ------------------------- K5: decoder GEMM (WMMA f32) ---------------------
// out(4096x8192) = H(4096x128) @ W_dec^T(128x8192) + b_dec. One wave per
// 16x16 tile; K=128 fully unrolled (32 wmma). Memory-bound: 134MB stores.
__global__ void __launch_bounds__(128) k_dec_gemm_wmma(
    const float* __restrict__ Hs, const float* __restrict__ Wdec,
    const float* __restrict__ bdec, float* __restrict__ out)
{
    const int wave = threadIdx.x >> 5;
    const int l    = threadIdx.x & 31;
    const int tn = (blockIdx.x * 4 + wave) * 16; // col tile base (0..8176)
    const int tm = blockIdx.y * 16;              // row tile base (0..4080)
    const int mr = l & 15;
    const int kg = (l >> 4) * 2;

    v8f acc = {};
    #pragma unroll
    for (int kb = 0; kb < HIDDEN; kb += 4) {
        // A: H[tm+mr, kb+kg+{0,1}]      -- contiguous pair
        v2f a = *(const v2f*)(Hs + (size_t)(tm + mr) * HIDDEN + kb + kg);
        // B: B[k][n] = Wdec[tn+mr, kb+kg+{0,1}] -- contiguous pair (Wdec row-major)
        v2f b = *(const v2f*)(Wdec + (size_t)(tn + mr) * HIDDEN + kb + kg);
        acc = __builtin_amdgcn_wmma_f32_16x16x4_f32(false, a, false, b,
                                                    (short)0, acc, false, false);
    }
    const float bias = bdec[tn + (l & 15)];
    const int mh = (l >> 4) * 8;
    #pragma unroll
    for (int v = 0; v < 8; ++v)
        out[(size_t)(tm + mh + v) * INPUT_DIM + tn + (l & 15)] = acc[v] + bias;
}

// ---------------------------------------------------------------------------
extern "C" void kernel_launch(void* const* d_in, const int* in_sizes, int n_in,
                              void* d_out, int out_size, void* d_ws, size_t ws_size,
                              hipStream_t stream) {
    const float* x0     = (const float*)d_in[0];
    const float* Wenc   = (const float*)d_in[1];
    const float* benc   = (const float*)d_in[2];
    const float* Wih    = (const float*)d_in[3];
    const float* Whh    = (const float*)d_in[4];
    const float* bias   = (const float*)d_in[5];
    const float* bias_n = (const float*)d_in[6];
    const float* Wdec   = (const float*)d_in[7];
    const float* bdec   = (const float*)d_in[8];
    float* out = (float*)d_out;

    // Workspace layout (floats): e0[128] c[128] M[16384] A[49152] u[384]
    //                            ig0[384] H[4096*128]   (~2.35 MB total)
    float* e0  = (float*)d_ws;
    float* c   = e0 + HIDDEN;
    float* M   = c + HIDDEN;
    float* A   = M + HIDDEN * HIDDEN;
    float* u   = A + GATES * HIDDEN;
    float* ig0 = u + GATES;
    float* Hs  = ig0 + GATES;

    k_enc_vec<<<HIDDEN, 256, 0, stream>>>(Wenc, benc, x0, bdec, e0, c);
    k_gemm_M_wmma<<<64, 32, 0, stream>>>(Wenc, Wdec, M);
    k_fold<<<GATES, HIDDEN, 0, stream>>>(Wih, bias, M, c, e0, A, u, ig0);
    k_gru_recur<<<1, ROWS, 0, stream>>>(A, Whh, u, ig0, bias_n, Hs);

    dim3 grid(INPUT_DIM / (4 * 16), STEPS / 16);   // (128, 256), 4 waves/block
    k_dec_gemm_wmma<<<grid, 128, 0, stream>>>(Hs, Wdec, bdec, out);
}